// GraphEncoder_15633680957569
// MI455X (gfx1250) — compile-verified
//
#include <hip/hip_runtime.h>

// GCN encoder for MI455X (gfx1250, wave32).
//   Per layer: XW = X @ W  (V_WMMA_F32_16X16X4_F32, fp32 exact; X tile staged
//              into padded LDS by the Tensor Data Mover)
//              ACC = scatter_add(norm * gather(XW, row), col)   (global_atomic_add_f32)
//              X'  = LayerNorm(ReLU(ACC + selfloop + b) + X)    (wave32 shuffle reduce)
// D = 128 fixed. N is a multiple of 16 (100000), E = 1.6M.

#define DD 128

typedef __attribute__((ext_vector_type(2))) float v2f;
typedef __attribute__((ext_vector_type(8))) float v8f;

__device__ __forceinline__ void atomic_add_f32(float* p, float v) {
#if defined(__HIP_DEVICE_COMPILE__)
  unsafeAtomicAdd(p, v);   // -> global_atomic_add_f32 on gfx1250
#else
  atomicAdd(p, v);
#endif
}

// ---------- degree kernels ----------
__global__ void k_deg_init(float* __restrict__ deg, int n) {
  int i = blockIdx.x * blockDim.x + threadIdx.x;
  if (i < n) deg[i] = 1.0f;   // self-loop
}

__global__ void k_deg_count(const long long* __restrict__ col, float* __restrict__ deg, int E) {
  int e = blockIdx.x * blockDim.x + threadIdx.x;
  if (e < E) atomic_add_f32(&deg[(int)col[e]], 1.0f);
}

__global__ void k_deg_rsqrt(float* __restrict__ deg, int n) {
  int i = blockIdx.x * blockDim.x + threadIdx.x;
  if (i < n) deg[i] = rsqrtf(deg[i]);
}

// ---------- zero accumulator ----------
__global__ void k_zero(float4* __restrict__ p, size_t n4) {
  size_t i = (size_t)blockIdx.x * blockDim.x + threadIdx.x;
  size_t stride = (size_t)gridDim.x * blockDim.x;
  float4 z = {0.f, 0.f, 0.f, 0.f};
  for (; i < n4; i += stride) p[i] = z;
}

// ---------- Tensor Data Mover: 16x128 f32 tile -> LDS with 4-DWORD row padding ----
#define LXS 132  // padded LDS row stride (floats): 132 % 64 = 4 -> conflict-free A reads

#if defined(__HIP_DEVICE_COMPILE__) && __has_builtin(__builtin_amdgcn_tensor_load_to_lds)
#define USE_TDM 1
typedef unsigned int v4u __attribute__((ext_vector_type(4)));
typedef int v4i __attribute__((ext_vector_type(4)));
typedef int v8i __attribute__((ext_vector_type(8)));

__device__ __forceinline__ void tdm_load_tile_16x128(const float* gsrc, void* lds_dst) {
  unsigned long long ga = (unsigned long long)(uintptr_t)gsrc;   // global byte address
  unsigned int lofs = (unsigned int)(uintptr_t)lds_dst;          // LDS byte offset = addr[31:0]
  // D# group 0 (ISA §8.3): count=1 | lds_addr | global_addr[56:0] | type=2
  v4u g0;
  g0[0] = 1u;
  g0[1] = lofs;
  g0[2] = (unsigned int)(ga & 0xFFFFFFFFu);
  g0[3] = (unsigned int)((ga >> 32) & 0x01FFFFFFu) | (2u << 30);
  // D# group 1 (ISA §8.4): data_size=2 (4B), pad_enable=1,
  // pad_interval=6 (pad every 128 DWORDs), pad_amount=3 (4 DWORDs) -> row stride 132 f32.
  v8i g1;
  g1[0] = (2 << 16) | (1 << 20) | (6 << 22) | (3 << 25);
  g1[1] = (int)(128u << 16);   // tensor_dim0[15:0]=128 (bits 79:48)
  g1[2] = (int)(16u << 16);    // tensor_dim0 hi=0 ; tensor_dim1[15:0]=16 (bits 111:80)
  g1[3] = (int)(128u << 16);   // tensor_dim1 hi=0 ; tile_dim0=128 (bits 127:112)
  g1[4] = 16;                  // tile_dim1=16, tile_dim2=0
  g1[5] = 128;                 // tensor_dim0_stride=128 elements (bits 207:160)
  g1[6] = 0;                   // stride hi / tensor_dim1_stride lo (unused, 2D)
  g1[7] = 0;
  v4i gz = {0, 0, 0, 0};       // groups 2/3: unused for 2D tile
#if __clang_major__ >= 23
  v8i gz8 = {0, 0, 0, 0, 0, 0, 0, 0};
  __builtin_amdgcn_tensor_load_to_lds(g0, g1, gz, gz, gz8, 0);
#else
  __builtin_amdgcn_tensor_load_to_lds(g0, g1, gz, gz, 0);
#endif
}
#else
#define USE_TDM 0
#endif

// ---------- GEMM: XW[N,128] = X[N,128] @ W[128,128] via WMMA f32 16x16x4 ----------
// Block = 256 threads = 8 waves. Block handles a 16-row strip; wave w handles
// the 16-column tile n0 = 16*w. 32 accumulating WMMAs over K.
__global__ void k_gemm(const float* __restrict__ X, const float* __restrict__ W,
                       float* __restrict__ XW) {
  __shared__ float lx[16 * LXS];
  const int m0  = blockIdx.x * 16;
  const int tid = threadIdx.x;

#if USE_TDM
  if (tid < 32) {  // wave 0 issues the DMA and waits on TENSORcnt
    tdm_load_tile_16x128(X + (size_t)m0 * DD, (void*)lx);
    __builtin_amdgcn_s_wait_tensorcnt(0);
  }
  __syncthreads();
#else
  for (int idx = tid; idx < 16 * DD; idx += 256) {
    int r = idx >> 7, c = idx & 127;
    lx[r * LXS + c] = X[(size_t)(m0 + r) * DD + c];
  }
  __syncthreads();
#endif

  const int wave = tid >> 5;
  const int lane = tid & 31;
  const int half = lane >> 4;   // 0: K pair (k0,k0+1), 1: K pair (k0+2,k0+3)
  const int l16  = lane & 15;
  const int n0   = wave * 16;

  v8f c = {};
  #pragma unroll 4
  for (int k0 = 0; k0 < DD; k0 += 4) {
    const int ka = k0 + half * 2;
    v2f a, b;
    // A 16x4 f32 (ISA layout): lanes 0-15 -> M=l16, K=k0,k0+1 ; lanes 16-31 -> K=k0+2,k0+3
    a.x = lx[l16 * LXS + ka];
    a.y = lx[l16 * LXS + ka + 1];
    // B 4x16 f32: row K striped across lanes (N = l16); halves hold K pairs like A
    b.x = W[(size_t)ka * DD + n0 + l16];
    b.y = W[(size_t)(ka + 1) * DD + n0 + l16];
    c = __builtin_amdgcn_wmma_f32_16x16x4_f32(false, a, false, b, (short)0, c, false, false);
  }

  // C/D 16x16 f32: lane<16 -> N=lane, rows r ; lane>=16 -> rows +8
  const int mb = m0 + half * 8;
  #pragma unroll
  for (int r = 0; r < 8; ++r)
    XW[(size_t)(mb + r) * DD + n0 + l16] = c[r];
}

// ---------- edge scatter: one wave per edge ----------
// lane loads float4 (32 lanes x 16B = 512B = full 128-f32 row), scales, atomically adds.
__global__ void k_scatter(const float* __restrict__ xw,
                          const long long* __restrict__ row,
                          const long long* __restrict__ col,
                          const float* __restrict__ dis,
                          float* __restrict__ acc, int E) {
  int wid = blockIdx.x * (blockDim.x >> 5) + (threadIdx.x >> 5);
#if defined(__HIP_DEVICE_COMPILE__)
  wid = __builtin_amdgcn_readfirstlane(wid);   // wave-uniform -> scalar loads for indices
#endif
  const int lane = threadIdx.x & 31;
  if (wid >= E) return;
  const int r    = (int)row[wid];
  const int cdst = (int)col[wid];
  const float nrm = dis[r] * dis[cdst];

  const float4 v = ((const float4*)(xw + (size_t)r * DD))[lane];
  float* dst = acc + (size_t)cdst * DD + lane * 4;
  atomic_add_f32(dst + 0, v.x * nrm);
  atomic_add_f32(dst + 1, v.y * nrm);
  atomic_add_f32(dst + 2, v.z * nrm);
  atomic_add_f32(dst + 3, v.w * nrm);
}

// ---------- fused epilogue: self-loop + bias + ReLU + residual + LayerNorm ----------
// one wave per node; each lane owns 4 features; mean/var via wave32 xor-shuffles.
__global__ void k_epilogue(const float* __restrict__ acc,
                           const float* __restrict__ xw,
                           const float* __restrict__ xin,
                           const float* __restrict__ dis,
                           const float* __restrict__ bias,
                           const float* __restrict__ lnw,
                           const float* __restrict__ lnb,
                           float* __restrict__ out, int n) {
  const int node = blockIdx.x * (blockDim.x >> 5) + (threadIdx.x >> 5);
  const int lane = threadIdx.x & 31;
  if (node >= n) return;

  const float ds = dis[node];
  const float sl = ds * ds;  // self-loop norm: dis[i]*dis[i]
  const size_t base = (size_t)node * DD + lane * 4;

  const float4 a  = *(const float4*)(acc + base);
  const float4 w  = *(const float4*)(xw + base);
  const float4 xi = *(const float4*)(xin + base);
  const float4 bb = *(const float4*)(bias + lane * 4);
  const float4 gw = *(const float4*)(lnw + lane * 4);
  const float4 gb = *(const float4*)(lnb + lane * 4);

  float h[4];
  h[0] = fmaxf(a.x + w.x * sl + bb.x, 0.f) + xi.x;
  h[1] = fmaxf(a.y + w.y * sl + bb.y, 0.f) + xi.y;
  h[2] = fmaxf(a.z + w.z * sl + bb.z, 0.f) + xi.z;
  h[3] = fmaxf(a.w + w.w * sl + bb.w, 0.f) + xi.w;

  float s = h[0] + h[1] + h[2] + h[3];
  #pragma unroll
  for (int o = 16; o > 0; o >>= 1) s += __shfl_xor(s, o, 32);
  const float mu = s * (1.0f / DD);

  float d0 = h[0] - mu, d1 = h[1] - mu, d2 = h[2] - mu, d3 = h[3] - mu;
  float q = d0 * d0 + d1 * d1 + d2 * d2 + d3 * d3;
  #pragma unroll
  for (int o = 16; o > 0; o >>= 1) q += __shfl_xor(q, o, 32);
  const float rs = rsqrtf(q * (1.0f / DD) + 1e-5f);

  float4 r;
  r.x = d0 * rs * gw.x + gb.x;
  r.y = d1 * rs * gw.y + gb.y;
  r.z = d2 * rs * gw.z + gb.z;
  r.w = d3 * rs * gw.w + gb.w;
  *(float4*)(out + base) = r;
}

extern "C" void kernel_launch(void* const* d_in, const int* in_sizes, int n_in,
                              void* d_out, int out_size, void* d_ws, size_t ws_size,
                              hipStream_t stream) {
  const float*     x   = (const float*)d_in[0];
  const long long* ei  = (const long long*)d_in[1];   // int64 [2, E]
  const float*     W   = (const float*)d_in[2];       // [3,128,128]
  const float*     b   = (const float*)d_in[3];       // [3,128]
  const float*     lnw = (const float*)d_in[4];
  const float*     lnb = (const float*)d_in[5];
  float*           out = (float*)d_out;

  const int N = in_sizes[0] / DD;   // 100000 (multiple of 16)
  const int E = in_sizes[1] / 2;    // 1600000
  const long long* row = ei;
  const long long* col = ei + E;

  // workspace layout
  char* ws = (char*)d_ws;
  const size_t nd_bytes = (size_t)N * DD * sizeof(float);
  float* dis = (float*)ws;
  size_t off = (((size_t)N * sizeof(float)) + 255) & ~(size_t)255;
  float* XW  = (float*)(ws + off); off += nd_bytes;
  float* ACC = (float*)(ws + off); off += nd_bytes;
  float* XA  = (float*)(ws + off);

  // degree -> deg^{-1/2} (self-loops included via init=1)
  k_deg_init <<<(N + 255) / 256, 256, 0, stream>>>(dis, N);
  k_deg_count<<<(E + 255) / 256, 256, 0, stream>>>(col, dis, E);
  k_deg_rsqrt<<<(N + 255) / 256, 256, 0, stream>>>(dis, N);

  const size_t nd4 = (size_t)N * DD / 4;
  const float* xc = x;
  for (int l = 0; l < 3; ++l) {
    k_gemm<<<N / 16, 256, 0, stream>>>(xc, W + (size_t)l * DD * DD, XW);
    k_zero<<<2048, 256, 0, stream>>>((float4*)ACC, nd4);
    k_scatter<<<(E + 7) / 8, 256, 0, stream>>>(XW, row, col, dis, ACC, E);
    float* xo = (l == 0) ? XA : out;   // L0 -> XA, L1 -> out, L2 -> out (in-place safe)
    k_epilogue<<<(N * 32 + 255) / 256, 256, 0, stream>>>(
        ACC, XW, xc, dis, b + (size_t)l * DD, lnw + (size_t)l * DD, lnb + (size_t)l * DD, xo, N);
    xc = xo;
  }
}